// UniformNormLossFunc_2_84490596646977
// MI455X (gfx1250) — compile-verified
//
#include <hip/hip_runtime.h>
#include <math.h>

// -------- CDNA5 (gfx1250) streaming L2-norm reduction ---------------------
// inputs: [N,N,D] f32 (N=1024, D=128), targets: [N,D] f32
// out    = sum_{j,i} || inputs[j,i,:] - targets[i,:] ||_2 / N   (scalar f32)
//
// HBM-bound: 512 MB single-use stream -> ~22 us floor at 23.3 TB/s.
// Strategy: one wave processes a tile of 16 contiguous rows; the per-row
// sum of squared diffs is accumulated with V_WMMA_F32_16X16X4_F32 against a
// ones B-matrix (full f32 accuracy). Inputs use non-temporal loads so the
// 512 KB targets array stays resident in the 192 MB L2.

typedef float v2f __attribute__((ext_vector_type(2)));
typedef float v4f __attribute__((ext_vector_type(4)));
typedef float v8f __attribute__((ext_vector_type(8)));

#ifndef __has_builtin
#define __has_builtin(x) 0
#endif

#if defined(__AMDGCN__) && __has_builtin(__builtin_amdgcn_wmma_f32_16x16x4_f32)
#define USE_WMMA_F32 1
#else
#define USE_WMMA_F32 0
#endif

constexpr int kN = 1024;
constexpr int kD = 128;
constexpr int kRows = kN * kN;          // 1,048,576 rows
constexpr int kTiles = kRows / 16;      // 65,536 tiles of 16 rows
constexpr int kBlocks = 2048;
constexpr int kThreads = 256;           // 8 waves per block (wave32)
constexpr int kWaves = kBlocks * (kThreads / 32);   // 16,384 waves

__global__ __launch_bounds__(kThreads)
void norm_sum_kernel(const float* __restrict__ in,
                     const float* __restrict__ tgt,
                     float* __restrict__ partial) {
  const int tid = threadIdx.x;
  const int lane = tid & 31;
  const int wave = blockIdx.x * (kThreads / 32) + (tid >> 5);

  float acc = 0.0f;

#if USE_WMMA_F32
  // 16x4 f32 A-matrix layout (ISA 7.12.2): lanes 0-15 hold row M=lane,
  // VGPR0=K0/VGPR1=K1; lanes 16-31 hold VGPR0=K2/VGPR1=K3.
  const int m  = lane & 15;          // row within tile
  const int kh = lane >> 4;          // K half (0 -> K0..1, 1 -> K2..3)
  const v2f ones = {1.0f, 1.0f};     // B = all-ones 4x16 (layout-invariant)

  for (int tile = wave; tile < kTiles; tile += kWaves) {   // uniform trip count
    const int R = tile * 16;                               // 16 contiguous rows
    const float* ip = in  + (size_t)(R + m) * kD + 2 * kh;
    const float* tp = tgt + (size_t)((R + m) & (kN - 1)) * kD + 2 * kh;

    v8f c0 = {};   // even k-chunks
    v8f c1 = {};   // odd  k-chunks (breaks the WMMA dependency chain)
#pragma unroll
    for (int c = 0; c < 32; c += 2) {            // k-chunks of 4 over D=128
      v2f x0 = __builtin_nontemporal_load((const v2f*)(ip + 4 * c));
      v2f x1 = __builtin_nontemporal_load((const v2f*)(ip + 4 * c + 4));
      v2f t0 = *(const v2f*)(tp + 4 * c);
      v2f t1 = *(const v2f*)(tp + 4 * c + 4);
      v2f d0 = x0 - t0;
      v2f d1 = x1 - t1;
      v2f a0 = d0 * d0;
      v2f a1 = d1 * d1;
      // C[m][n] += sum_k A[m][k] : row-sum via ones-matrix, f32 exact path
      c0 = __builtin_amdgcn_wmma_f32_16x16x4_f32(false, a0, false, ones,
                                                 (short)0, c0, false, false);
      c1 = __builtin_amdgcn_wmma_f32_16x16x4_f32(false, a1, false, ones,
                                                 (short)0, c1, false, false);
    }
    // Each lane now holds 8 row-sums (rows replicated across the 16 columns).
    // All lanes sqrt uniformly (keeps EXEC all-ones); 16x overcount is
    // compensated below.
#pragma unroll
    for (int v = 0; v < 8; ++v) acc += sqrtf(c0[v] + c1[v]);
  }
  const float waveScale = 1.0f / 16.0f;   // column replication factor
#else
  // Fallback: one row per wave, lane covers 4 floats, butterfly reduce.
  for (int r = wave; r < kRows; r += kWaves) {   // uniform trip count
    const int i = r & (kN - 1);
    v4f x = __builtin_nontemporal_load(
        (const v4f*)(in + (size_t)r * kD + lane * 4));
    v4f t = *(const v4f*)(tgt + (size_t)i * kD + lane * 4);
    v4f d = x - t;
    float p = d[0] * d[0] + d[1] * d[1] + d[2] * d[2] + d[3] * d[3];
#pragma unroll
    for (int off = 16; off > 0; off >>= 1) p += __shfl_xor(p, off, 32);
    acc += sqrtf(p);                       // all 32 lanes add -> 32x overcount
  }
  const float waveScale = 1.0f / 32.0f;
#endif

  // Deterministic block reduction.
  __shared__ float red[kThreads];
  red[tid] = acc;
  __syncthreads();
#pragma unroll
  for (int s = kThreads / 2; s > 0; s >>= 1) {
    if (tid < s) red[tid] += red[tid + s];
    __syncthreads();
  }
  if (tid == 0) partial[blockIdx.x] = red[0] * waveScale;
}

__global__ __launch_bounds__(256)
void final_reduce_kernel(const float* __restrict__ partial,
                         float* __restrict__ out, int n) {
  const int tid = threadIdx.x;
  float s = 0.0f;
  for (int i = tid; i < n; i += 256) s += partial[i];   // fixed order
  __shared__ float red[256];
  red[tid] = s;
  __syncthreads();
#pragma unroll
  for (int k = 128; k > 0; k >>= 1) {
    if (tid < k) red[tid] += red[tid + k];
    __syncthreads();
  }
  if (tid == 0) out[0] = red[0] * (1.0f / (float)kN);
}

extern "C" void kernel_launch(void* const* d_in, const int* in_sizes, int n_in,
                              void* d_out, int out_size, void* d_ws,
                              size_t ws_size, hipStream_t stream) {
  const float* in  = (const float*)d_in[0];   // [N,N,D] f32
  const float* tgt = (const float*)d_in[1];   // [N,D]   f32
  float* out = (float*)d_out;                 // scalar f32
  float* partial = (float*)d_ws;              // kBlocks floats of scratch

  norm_sum_kernel<<<kBlocks, kThreads, 0, stream>>>(in, tgt, partial);
  final_reduce_kernel<<<1, 256, 0, stream>>>(partial, out, kBlocks);
}